// CapsuleLayer_81226421502346
// MI455X (gfx1250) — compile-verified
//
#include <hip/hip_runtime.h>

typedef float v2f __attribute__((ext_vector_type(2)));
typedef float v8f __attribute__((ext_vector_type(8)));

namespace {
constexpr int kInCaps   = 1152;
constexpr int kInDim    = 8;
constexpr int kOutCaps  = 10;
constexpr int kOutDim   = 16;
constexpr int kNumWaves = 8;
constexpr int kThreads  = kNumWaves * 32;   // 256 threads = 8 wave32
constexpr float kEps    = 1e-7f;
}

// One block handles a tile of 16 batch elements. u_hat is recomputed per
// routing pass via V_WMMA_F32_16X16X4_F32 (two K=4 steps cover K=8), so the
// 189 MB u_hat tensor never touches HBM. Routing logits use the linearity
// b_k = (v0+..+v_{k-1})·u_hat, so only a running vsum lives in LDS.
__launch_bounds__(kThreads)
__global__ void capsule_routing_wmma(const float* __restrict__ x,
                                     const float* __restrict__ W,
                                     float* __restrict__ out)
{
  __shared__ float s_red[kOutCaps * 16 * 16]; // pre-squash s[j][b][d]
  __shared__ float s_v[kOutCaps * 16 * 16];   // running sum of squashed outputs

  const int tid  = threadIdx.x;
  const int wave = tid >> 5;
  const int lane = tid & 31;
  const int m    = lane & 15;   // A: row M / B,C,D: column N (= d)
  const int hi   = lane >> 4;   // half-wave select (K pair / M offset)
  const int b0   = blockIdx.x * 16;

  for (int idx = tid; idx < kOutCaps * 256; idx += kThreads) s_v[idx] = 0.f;
  __syncthreads();

  for (int pass = 0; pass < 3; ++pass) {
    // clear reduction buffer (consumed after the i-loop)
    for (int idx = tid; idx < kOutCaps * 256; idx += kThreads) s_red[idx] = 0.f;

    // replicate vsum into the WMMA C/D register layout: vreg[j][r] holds
    // vsum[b = r + 8*hi][j][d = m]
    float vreg[kOutCaps][8];
    if (pass > 0) {
#pragma unroll
      for (int j = 0; j < kOutCaps; ++j)
#pragma unroll
        for (int r = 0; r < 8; ++r)
          vreg[j][r] = s_v[(j * 16 + (r + 8 * hi)) * 16 + m];
    }
    __syncthreads();

    float acc[kOutCaps][8];
#pragma unroll
    for (int j = 0; j < kOutCaps; ++j)
#pragma unroll
      for (int r = 0; r < 8; ++r) acc[j][r] = 0.f;

    for (int i = wave; i < kInCaps; i += kNumWaves) {
      // A tile (16 batches x 8 K): lanes 0-15 hold K=0,1 ; lanes 16-31 K=2,3
      const float* xp = x + ((size_t)(b0 + m) * kInCaps + i) * kInDim + 2 * hi;
      v2f a0, a1;
      a0[0] = xp[0]; a0[1] = xp[1];   // K = 2*hi, 2*hi+1
      a1[0] = xp[4]; a1[1] = xp[5];   // K = 4+2*hi, 5+2*hi

      float u[kOutCaps][8];           // u_hat tiles for all 10 output caps
#pragma unroll
      for (int j = 0; j < kOutCaps; ++j) {
        const float* wp =
            W + (((size_t)i * kOutCaps + j) * kInDim + 2 * hi) * kOutDim + m;
        v2f bm0, bm1;
        bm0[0] = wp[0];                // K = 2*hi      , N = m
        bm0[1] = wp[kOutDim];          // K = 2*hi + 1
        bm1[0] = wp[4 * kOutDim];      // K = 2*hi + 4
        bm1[1] = wp[5 * kOutDim];      // K = 2*hi + 5
        v8f c = {};
        c = __builtin_amdgcn_wmma_f32_16x16x4_f32(false, a0, false, bm0,
                                                  (short)0, c, false, false);
        c = __builtin_amdgcn_wmma_f32_16x16x4_f32(false, a1, false, bm1,
                                                  (short)0, c, false, false);
#pragma unroll
        for (int r = 0; r < 8; ++r) u[j][r] = c[r];
      }

      if (pass == 0) {
        // softmax(0) is uniform: accumulate raw, scale by 1/10 at the end
#pragma unroll
        for (int j = 0; j < kOutCaps; ++j)
#pragma unroll
          for (int r = 0; r < 8; ++r) acc[j][r] += u[j][r];
      } else {
#pragma unroll
        for (int r = 0; r < 8; ++r) {
          // logit[b,i,j] = vsum[b,j,:] . u_hat[b,i,j,:]  (reduce over d=lanes)
          float logit[kOutCaps];
#pragma unroll
          for (int j = 0; j < kOutCaps; ++j) {
            float t = vreg[j][r] * u[j][r];
            t += __shfl_xor(t, 1, 32);
            t += __shfl_xor(t, 2, 32);
            t += __shfl_xor(t, 4, 32);
            t += __shfl_xor(t, 8, 32);   // masks < 16: stays in 16-lane half
            logit[j] = t;
          }
          float mx = logit[0];
#pragma unroll
          for (int j = 1; j < kOutCaps; ++j) mx = fmaxf(mx, logit[j]);
          float e[kOutCaps];
          float se = 0.f;
#pragma unroll
          for (int j = 0; j < kOutCaps; ++j) {
            e[j] = __expf(logit[j] - mx);
            se += e[j];
          }
          const float inv = 1.f / se;
#pragma unroll
          for (int j = 0; j < kOutCaps; ++j)
            acc[j][r] += (e[j] * inv) * u[j][r];
        }
      }
    }

    // Deterministic cross-wave reduction: waves add into LDS one at a time.
    const float cscale = (pass == 0) ? (1.f / kOutCaps) : 1.f;
    for (int w = 0; w < kNumWaves; ++w) {
      if (wave == w) {
#pragma unroll
        for (int j = 0; j < kOutCaps; ++j)
#pragma unroll
          for (int r = 0; r < 8; ++r)
            s_red[(j * 16 + (r + 8 * hi)) * 16 + m] += acc[j][r] * cscale;
      }
      __syncthreads();
    }

    // squash: one thread per (j, b) pair
    if (tid < kOutCaps * 16) {
      const int j = tid >> 4;
      const int b = tid & 15;
      const float* sp = &s_red[(j * 16 + b) * 16];
      float sv[kOutDim];
      float s2 = 0.f;
#pragma unroll
      for (int d = 0; d < kOutDim; ++d) {
        sv[d] = sp[d];
        s2 += sv[d] * sv[d];
      }
      const float scale = s2 / (1.f + s2) / sqrtf(s2 + kEps);
#pragma unroll
      for (int d = 0; d < kOutDim; ++d) {
        const float v = scale * sv[d];
        if (pass < 2)
          s_v[(j * 16 + b) * 16 + d] += v;   // vsum for next pass's logits
        else
          out[(((size_t)(b0 + b)) * kOutCaps + j) * kOutDim + d] = v;
      }
    }
    __syncthreads();
  }
}

extern "C" void kernel_launch(void* const* d_in, const int* in_sizes, int n_in,
                              void* d_out, int out_size, void* d_ws, size_t ws_size,
                              hipStream_t stream) {
  (void)in_sizes; (void)n_in; (void)out_size; (void)d_ws; (void)ws_size;
  const float* x = (const float*)d_in[0];
  const float* W = (const float*)d_in[1];
  float* out = (float*)d_out;
  dim3 grid(256 / 16);     // 16 batch tiles of 16
  dim3 block(kThreads);    // 8 wave32
  hipLaunchKernelGGL(capsule_routing_wmma, grid, block, 0, stream, x, W, out);
}